// DSR_GCN_30030411334238
// MI455X (gfx1250) — compile-verified
//
#include <hip/hip_runtime.h>
#include <math.h>

#define HW_TOT 65536
#define NBIG   1024
#define NSMALL 2048
#define NCLS   16

typedef __attribute__((ext_vector_type(2))) float v2f;
typedef __attribute__((ext_vector_type(8))) float v8f;

// ---------------------------------------------------------------------------
// fp32 WMMA: D(16x16,f32) = A(16x4,f32) * B(4x16,f32) + C
// A layout (ISA 7.12.2): lane L: m = L&15, holds K = 2*(L>>4) + {0,1}
// C/D layout: VGPR i: n = L&15, m = i + 8*(L>>4)
// ---------------------------------------------------------------------------
__device__ __forceinline__ v8f wmma_f32(v2f a, v2f b, v8f c) {
  return __builtin_amdgcn_wmma_f32_16x16x4_f32(
      /*neg_a=*/false, a, /*neg_b=*/false, b,
      /*c_mod=*/(short)0, c, /*reuse_a=*/false, /*reuse_b=*/false);
}

// ---------------------------------------------------------------------------
// Column sums (deterministic two-pass, no float atomics) -> inv = 1/colsum
// ---------------------------------------------------------------------------
__global__ void colsum_partial_kernel(const float* __restrict__ Q, int rows, int cols,
                                      int rpc, float* __restrict__ part) {
  int col = blockIdx.x * blockDim.x + threadIdx.x;
  if (col >= cols) return;
  int r0 = blockIdx.y * rpc;
  int r1 = r0 + rpc; if (r1 > rows) r1 = rows;
  float s = 0.f;
  for (int r = r0; r < r1; ++r) s += Q[(size_t)r * cols + col];
  part[(size_t)blockIdx.y * cols + col] = s;
}

__global__ void colsum_finish_kernel(const float* __restrict__ part, int chunks, int cols,
                                     float* __restrict__ inv) {
  int col = blockIdx.x * blockDim.x + threadIdx.x;
  if (col >= cols) return;
  float s = 0.f;
  for (int c = 0; c < chunks; ++c) s += part[(size_t)c * cols + col];
  inv[col] = 1.0f / s;
}

// ---------------------------------------------------------------------------
// H[m,n] = inv[m] * sum_k Q[k*M + m] * F[k*Nc + n]   (H = diag(inv) * Q^T F)
// one 16x16 output tile per wave; 8 waves / block
// ---------------------------------------------------------------------------
__global__ void __launch_bounds__(256)
gemm_qtf_kernel(const float* __restrict__ Q, const float* __restrict__ F,
                const float* __restrict__ inv, float* __restrict__ H,
                int M, int Nc, int K) {
  int lane = threadIdx.x & 31;
  int wave = threadIdx.x >> 5;
  int tiles_n = Nc >> 4;
  int tile = blockIdx.x * (blockDim.x >> 5) + wave;
  int m0 = (tile / tiles_n) << 4;
  int n0 = (tile % tiles_n) << 4;
  int mi = m0 + (lane & 15);
  int ni = n0 + (lane & 15);
  int kb = (lane >> 4) << 1;
  v8f acc = {};
#pragma unroll 4
  for (int k = 0; k < K; k += 4) {
    int kk = k + kb;
    v2f a, b;
    a.x = Q[(size_t)kk * M + mi];
    a.y = Q[(size_t)(kk + 1) * M + mi];
    b.x = F[(size_t)kk * Nc + ni];
    b.y = F[(size_t)(kk + 1) * Nc + ni];
    acc = wmma_f32(a, b, acc);
  }
  int half = lane >> 4;
#pragma unroll
  for (int i = 0; i < 8; ++i) {
    int m = m0 + i + 8 * half;
    H[(size_t)m * Nc + ni] = inv[m] * acc[i];
  }
}

// ---------------------------------------------------------------------------
// BatchNorm1d (training): per-column mean / rsqrt(var+eps) over N rows
// ---------------------------------------------------------------------------
__global__ void bn_stats_kernel(const float* __restrict__ H, int N, int ind,
                                float* __restrict__ stats) {
  int j = blockIdx.x;
  __shared__ float ssum[256], ssq[256];
  float s = 0.f, q = 0.f;
  for (int r = threadIdx.x; r < N; r += blockDim.x) {
    float v = H[(size_t)r * ind + j];
    s += v; q += v * v;
  }
  ssum[threadIdx.x] = s; ssq[threadIdx.x] = q;
  __syncthreads();
  for (int st = blockDim.x >> 1; st > 0; st >>= 1) {
    if (threadIdx.x < (unsigned)st) {
      ssum[threadIdx.x] += ssum[threadIdx.x + st];
      ssq[threadIdx.x]  += ssq[threadIdx.x + st];
    }
    __syncthreads();
  }
  if (threadIdx.x == 0) {
    float mean = ssum[0] / (float)N;
    float var  = ssq[0] / (float)N - mean * mean;
    stats[j] = mean;
    stats[ind + j] = rsqrtf(var + 1e-5f);
  }
}

__global__ void bn_apply_kernel(const float* __restrict__ H, int N, int ind,
                                const float* __restrict__ stats,
                                const float* __restrict__ g, const float* __restrict__ b,
                                float* __restrict__ out) {
  int idx = blockIdx.x * blockDim.x + threadIdx.x;
  if (idx >= N * ind) return;
  int j = idx % ind;
  out[idx] = (H[idx] - stats[j]) * stats[ind + j] * g[j] + b[j];
}

// ---------------------------------------------------------------------------
// C = A @ W^T (+bias) ; A:[M,K] rm, W:[Nc,K] rm.  Both fragments = contiguous
// float2 loads.  mode 0: C=acc+bias ; mode 1 (Gram->adjacency epilogue):
//   An = max(sigmoid(acc), clamp) * Adj + I    (clamp skipped if clampv<0)
// ---------------------------------------------------------------------------
__global__ void __launch_bounds__(256)
gemm_abt_kernel(const float* __restrict__ A, const float* __restrict__ W,
                const float* __restrict__ bias, float* __restrict__ Cout,
                int M, int Nc, int K, int ldc,
                int mode, const float* __restrict__ Adj, float clampv) {
  int lane = threadIdx.x & 31;
  int wave = threadIdx.x >> 5;
  int tiles_n = Nc >> 4;
  int tile = blockIdx.x * (blockDim.x >> 5) + wave;
  int m0 = (tile / tiles_n) << 4;
  int n0 = (tile % tiles_n) << 4;
  int nn = n0 + (lane & 15);
  int kb = (lane >> 4) << 1;
  const float* arow = A + (size_t)(m0 + (lane & 15)) * K;
  const float* brow = W + (size_t)nn * K;
  v8f acc = {};
#pragma unroll 4
  for (int k = 0; k < K; k += 4) {
    v2f a = *(const v2f*)(arow + k + kb);
    v2f b = *(const v2f*)(brow + k + kb);
    acc = wmma_f32(a, b, acc);
  }
  int half = lane >> 4;
#pragma unroll
  for (int i = 0; i < 8; ++i) {
    int m = m0 + i + 8 * half;
    float v = acc[i];
    if (mode == 0) {
      if (bias) v += bias[nn];
      Cout[(size_t)m * ldc + nn] = v;
    } else {
      float s = 1.0f / (1.0f + expf(-v));
      if (clampv >= 0.f) s = fmaxf(s, clampv);
      Cout[(size_t)m * ldc + nn] =
          s * Adj[(size_t)m * ldc + nn] + ((m == nn) ? 1.0f : 0.0f);
    }
  }
}

// ---------------------------------------------------------------------------
// dinv[row] = rsqrt( sum_j An[row, j] )
// ---------------------------------------------------------------------------
__global__ void rowsum_dinv_kernel(const float* __restrict__ An, int N,
                                   float* __restrict__ dinv) {
  int row = blockIdx.x;
  __shared__ float ssum[256];
  const float* r = An + (size_t)row * N;
  float s = 0.f;
  for (int j = threadIdx.x; j < N; j += blockDim.x) s += r[j];
  ssum[threadIdx.x] = s;
  __syncthreads();
  for (int st = blockDim.x >> 1; st > 0; st >>= 1) {
    if (threadIdx.x < (unsigned)st) ssum[threadIdx.x] += ssum[threadIdx.x + st];
    __syncthreads();
  }
  if (threadIdx.x == 0) dinv[row] = rsqrtf(ssum[0]);
}

// ---------------------------------------------------------------------------
// C = A @ B ; A:[M,K] rm (lda), B:[K,Nc] rm.
// mode 0: C = acc                       (res = Q @ H)
// mode 1: C = leakyrelu( dinv[m] * sum_k A[m,k]*dinv[k]*B[k,n] )  (A_hat prop)
// ---------------------------------------------------------------------------
__global__ void __launch_bounds__(256)
gemm_ab_kernel(const float* __restrict__ A, const float* __restrict__ B,
               float* __restrict__ Cout, int M, int Nc, int K, int lda,
               int mode, const float* __restrict__ dinv) {
  int lane = threadIdx.x & 31;
  int wave = threadIdx.x >> 5;
  int tiles_n = Nc >> 4;
  int tile = blockIdx.x * (blockDim.x >> 5) + wave;
  int m0 = (tile / tiles_n) << 4;
  int n0 = (tile % tiles_n) << 4;
  int ni = n0 + (lane & 15);
  int kb = (lane >> 4) << 1;
  const float* arow = A + (size_t)(m0 + (lane & 15)) * lda;
  v8f acc = {};
#pragma unroll 4
  for (int k = 0; k < K; k += 4) {
    int kk = k + kb;
    v2f a = *(const v2f*)(arow + kk);
    float b0 = B[(size_t)kk * Nc + ni];
    float b1 = B[(size_t)(kk + 1) * Nc + ni];
    if (mode == 1) { b0 *= dinv[kk]; b1 *= dinv[kk + 1]; }
    v2f b; b.x = b0; b.y = b1;
    acc = wmma_f32(a, b, acc);
  }
  int half = lane >> 4;
#pragma unroll
  for (int i = 0; i < 8; ++i) {
    int m = m0 + i + 8 * half;
    float v = acc[i];
    if (mode == 1) {
      v *= dinv[m];
      v = (v > 0.f) ? v : 0.01f * v;   // leaky_relu(0.01)
    }
    Cout[(size_t)m * Nc + ni] = v;
  }
}

// ---------------------------------------------------------------------------
// Fused head: logits/softmax + squared-difference loss.
//   z      = sigma*rb + (1-sigma)*rs  (first 64); concat tail is y in both.
//   loss_c = ((rb - rs) @ W64_c)^2   (lin64 biases cancel)
// ---------------------------------------------------------------------------
__global__ void __launch_bounds__(256)
head_kernel(const float* __restrict__ rb, const float* __restrict__ rs,
            const float* __restrict__ y,
            const float* __restrict__ W128, const float* __restrict__ b128,
            const float* __restrict__ W64,
            const float* __restrict__ sigma2,
            float* __restrict__ Yout, float* __restrict__ Lout) {
  __shared__ float sW128[NCLS * 128];
  __shared__ float sW64[NCLS * 64];
  __shared__ float sb128[NCLS];
  for (int i = threadIdx.x; i < NCLS * 128; i += blockDim.x) sW128[i] = W128[i];
  for (int i = threadIdx.x; i < NCLS * 64; i += blockDim.x) sW64[i] = W64[i];
  if (threadIdx.x < NCLS) sb128[threadIdx.x] = b128[threadIdx.x];
  __syncthreads();

  int p = blockIdx.x * blockDim.x + threadIdx.x;
  float sg = sigma2[0];
  float z[64], dif[64];
#pragma unroll
  for (int j = 0; j < 64; ++j) {
    float a = rb[(size_t)p * 64 + j];
    float b = rs[(size_t)p * 64 + j];
    z[j]   = sg * a + (1.0f - sg) * b;
    dif[j] = a - b;
  }
  float logits[NCLS], mx = -3.4e38f;
#pragma unroll
  for (int c = 0; c < NCLS; ++c) {
    float acc = sb128[c];
    const float* wr = &sW128[c * 128];
#pragma unroll
    for (int j = 0; j < 64; ++j) acc += z[j] * wr[j];
    for (int j = 0; j < 64; ++j) acc += y[(size_t)p * 64 + j] * wr[64 + j];
    logits[c] = acc;
    mx = fmaxf(mx, acc);
  }
  float sum = 0.f;
#pragma unroll
  for (int c = 0; c < NCLS; ++c) { logits[c] = expf(logits[c] - mx); sum += logits[c]; }
  float rinv = 1.0f / sum;
#pragma unroll
  for (int c = 0; c < NCLS; ++c) Yout[(size_t)p * NCLS + c] = logits[c] * rinv;
#pragma unroll
  for (int c = 0; c < NCLS; ++c) {
    float acc = 0.f;
    const float* wr = &sW64[c * 64];
#pragma unroll
    for (int j = 0; j < 64; ++j) acc += dif[j] * wr[j];
    Lout[(size_t)p * NCLS + c] = acc * acc;
  }
}

// ---------------------------------------------------------------------------
// One GCN layer: BN -> theta proj -> sigmoid Gram adjacency -> normalize ->
// propagate -> leaky relu
// ---------------------------------------------------------------------------
static void gcn_layer(hipStream_t stream, const float* Hin, float* Hout,
                      int N, int ind, int outd,
                      const float* bng, const float* bnb,
                      const float* thW, const float* thb,
                      const float* oW, const float* ob,
                      const float* Adj, float clampv,
                      float* Hbn, float* stats, float* Hx, float* An,
                      float* dinv, float* HoW) {
  bn_stats_kernel<<<ind, 256, 0, stream>>>(Hin, N, ind, stats);
  int tot = N * ind;
  bn_apply_kernel<<<(tot + 255) / 256, 256, 0, stream>>>(Hin, N, ind, stats, bng, bnb, Hbn);
  // Hx = Hbn @ thW^T + thb : [N, 256]
  gemm_abt_kernel<<<(N / 16) * (256 / 16) / 8, 256, 0, stream>>>(
      Hbn, thW, thb, Hx, N, 256, ind, 256, 0, nullptr, -1.f);
  // An = clamp(sigmoid(Hx @ Hx^T)) * Adj + I : [N, N]
  gemm_abt_kernel<<<(N / 16) * (N / 16) / 8, 256, 0, stream>>>(
      Hx, Hx, nullptr, An, N, N, 256, N, 1, Adj, clampv);
  rowsum_dinv_kernel<<<N, 256, 0, stream>>>(An, N, dinv);
  // HoW = Hbn @ oW^T + ob : [N, outd]
  gemm_abt_kernel<<<(N / 16) * (outd / 16) / 8, 256, 0, stream>>>(
      Hbn, oW, ob, HoW, N, outd, ind, outd, 0, nullptr, -1.f);
  // Hout = leakyrelu( D^-1/2 An D^-1/2 @ HoW )
  gemm_ab_kernel<<<(N / 16) * (outd / 16) / 8, 256, 0, stream>>>(
      An, HoW, Hout, N, outd, N, N, 1, dinv);
}

// ---------------------------------------------------------------------------
extern "C" void kernel_launch(void* const* d_in, const int* in_sizes, int n_in,
                              void* d_out, int out_size, void* d_ws, size_t ws_size,
                              hipStream_t stream) {
  const float* x    = (const float*)d_in[0];   // [HW, 128] (flat view)
  const float* y    = (const float*)d_in[1];   // [HW, 64]
  const float* Q    = (const float*)d_in[2];   // [HW, 1024]
  const float* Ab   = (const float*)d_in[3];   // [1024, 1024]
  const float* Qs   = (const float*)d_in[4];   // [HW, 2048]
  const float* As   = (const float*)d_in[5];   // [2048, 2048]
  const float* prm[24];
  for (int i = 0; i < 24; ++i) prm[i] = (const float*)d_in[6 + i];
  // per branch-layer: [bng, bnb, thW, thb, oW, ob]
  const float* const* b0 = prm + 0;
  const float* const* b1 = prm + 6;
  const float* const* s0 = prm + 12;
  const float* const* s1 = prm + 18;
  const float* lin128_W = (const float*)d_in[30];
  const float* lin128_b = (const float*)d_in[31];
  const float* lin64_W  = (const float*)d_in[32];
  const float* sigma2   = (const float*)d_in[34];

  float* Yout = (float*)d_out;
  float* Lout = Yout + (size_t)HW_TOT * NCLS;

  // ---- workspace carve (floats) ----
  float* ws = (float*)d_ws;
  size_t o = 0;
  float* inv_big   = ws + o; o += NBIG;
  float* inv_small = ws + o; o += NSMALL;
  float* part      = ws + o; o += (size_t)64 * NSMALL;
  float* bufA      = ws + o; o += (size_t)NSMALL * 128;
  float* bufB      = ws + o; o += (size_t)NSMALL * 128;
  float* Hbn       = ws + o; o += (size_t)NSMALL * 128;
  float* stats     = ws + o; o += 2 * 128;
  float* Hx        = ws + o; o += (size_t)NSMALL * 256;
  float* An        = ws + o; o += (size_t)NSMALL * NSMALL;
  float* dinv      = ws + o; o += NSMALL;
  float* HoW       = ws + o; o += (size_t)NSMALL * 128;
  float* res_big   = ws + o; o += (size_t)HW_TOT * 64;
  float* res_small = ws + o; o += (size_t)HW_TOT * 64;

  const int CHUNKS = 64, RPC = HW_TOT / CHUNKS;

  // ---- big branch ----
  colsum_partial_kernel<<<dim3(NBIG / 256, CHUNKS), 256, 0, stream>>>(Q, HW_TOT, NBIG, RPC, part);
  colsum_finish_kernel<<<NBIG / 256, 256, 0, stream>>>(part, CHUNKS, NBIG, inv_big);
  gemm_qtf_kernel<<<(NBIG / 16) * (128 / 16) / 8, 256, 0, stream>>>(
      Q, x, inv_big, bufA, NBIG, 128, HW_TOT);
  gcn_layer(stream, bufA, bufB, NBIG, 128, 128,
            b0[0], b0[1], b0[2], b0[3], b0[4], b0[5], Ab, 0.03f,
            Hbn, stats, Hx, An, dinv, HoW);
  gcn_layer(stream, bufB, bufA, NBIG, 128, 64,
            b1[0], b1[1], b1[2], b1[3], b1[4], b1[5], Ab, 0.03f,
            Hbn, stats, Hx, An, dinv, HoW);
  // res_big = Q @ H1 : [HW, 64]
  gemm_ab_kernel<<<(HW_TOT / 16) * (64 / 16) / 8, 256, 0, stream>>>(
      Q, bufA, res_big, HW_TOT, 64, NBIG, NBIG, 0, nullptr);

  // ---- small branch ----
  colsum_partial_kernel<<<dim3(NSMALL / 256, CHUNKS), 256, 0, stream>>>(Qs, HW_TOT, NSMALL, RPC, part);
  colsum_finish_kernel<<<NSMALL / 256, 256, 0, stream>>>(part, CHUNKS, NSMALL, inv_small);
  gemm_qtf_kernel<<<(NSMALL / 16) * (128 / 16) / 8, 256, 0, stream>>>(
      Qs, x, inv_small, bufA, NSMALL, 128, HW_TOT);
  gcn_layer(stream, bufA, bufB, NSMALL, 128, 128,
            s0[0], s0[1], s0[2], s0[3], s0[4], s0[5], As, -1.f,
            Hbn, stats, Hx, An, dinv, HoW);
  gcn_layer(stream, bufB, bufA, NSMALL, 128, 64,
            s1[0], s1[1], s1[2], s1[3], s1[4], s1[5], As, -1.f,
            Hbn, stats, Hx, An, dinv, HoW);
  gemm_ab_kernel<<<(HW_TOT / 16) * (64 / 16) / 8, 256, 0, stream>>>(
      Qs, bufA, res_small, HW_TOT, 64, NSMALL, NSMALL, 0, nullptr);

  // ---- fused head ----
  head_kernel<<<HW_TOT / 256, 256, 0, stream>>>(
      res_big, res_small, y, lin128_W, lin128_b, lin64_W, sigma2, Yout, Lout);
}